// VoronoiValues_38173669326936
// MI455X (gfx1250) — compile-verified
//
#include <hip/hip_runtime.h>
#include <hip/hip_bf16.h>
#include <math.h>

typedef __attribute__((ext_vector_type(2))) float v2f;
typedef __attribute__((ext_vector_type(4))) float v4f;
typedef __attribute__((ext_vector_type(8))) float v8f;

#define KNN            11
#define ROWS_PER_BLOCK 128
#define THREADS        128
#define CHUNK_COLS     64
#define LDS_STRIDE     68   // dwords; row = 272 B = 16B-aligned -> b128 scan loads

__global__ __launch_bounds__(THREADS)
void VoronoiValues_kernel(const float* __restrict__ points,
                          const float* __restrict__ cells,
                          float* __restrict__ out,
                          int M, int N) {
  __shared__ __align__(16) float S[ROWS_PER_BLOCK * LDS_STRIDE];   // 34,816 B

  const int tid  = threadIdx.x;
  const int lane = tid & 31;
  const int wave = tid >> 5;
  const int half = lane >> 4;   // 0: lanes 0-15, 1: lanes 16-31
  const int mrow = lane & 15;

  // ---- A tiles (fixed per wave): 16x4 f32, row m = (-2px,-2py,-2pz,1) ----
  // Layout (ISA 7.12.2): lanes 0-15 hold K=0 (VGPR0) / K=1 (VGPR1),
  //                      lanes 16-31 hold K=2 (VGPR0) / K=3 (VGPR1).
  const int rowBase = blockIdx.x * ROWS_PER_BLOCK + wave * 32;
  v2f a[2];
#pragma unroll
  for (int t = 0; t < 2; ++t) {
    const int r = rowBase + t * 16 + mrow;
    const float px = points[3*r+0], py = points[3*r+1], pz = points[3*r+2];
    a[t].x = half ? (-2.0f * pz) : (-2.0f * px);
    a[t].y = half ? 1.0f         : (-2.0f * py);
  }

  // ---- per-thread top-K (ascending; best[0] = nearest) ----
  float best[KNN];
  int   bidx[KNN];
#pragma unroll
  for (int k = 0; k < KNN; ++k) { best[k] = 3.4e38f; bidx[k] = 0; }

  const v4f* Srow = (const v4f*)&S[tid * LDS_STRIDE];

  for (int j0 = 0; j0 < N; j0 += CHUNK_COLS) {
    // ---- Stage A: WMMA score tiles -> LDS ----
    // score[i][j] = -2 p_i . c_j + |c_j|^2  (row-constant |p_i|^2 dropped:
    // per-row top-k ordering is unchanged)
#pragma unroll
    for (int ct = 0; ct < CHUNK_COLS / 16; ++ct) {
      const int j  = j0 + ct * 16 + mrow;
      const float cx = cells[3*j+0], cy = cells[3*j+1], cz = cells[3*j+2];
      const float n2 = cx*cx + cy*cy + cz*cz;
      // B 4x16 f32: col n = (cx,cy,cz,|c|^2); lanes 0-15: K=0/1, lanes 16-31: K=2/3.
      v2f b;
      b.x = half ? cz : cx;
      b.y = half ? n2 : cy;
#pragma unroll
      for (int t = 0; t < 2; ++t) {
        v8f c = {};
        c = __builtin_amdgcn_wmma_f32_16x16x4_f32(
                /*neg_a=*/false, a[t], /*neg_b=*/false, b,
                /*c_mod=*/(short)0, c, /*reuse_a=*/false, /*reuse_b=*/false);
        // C/D layout: VGPR v -> row v (lanes 0-15) or v+8 (lanes 16-31), col = lane%16
        const int rloc = wave * 32 + t * 16 + half * 8;
        const int cloc = ct * 16 + mrow;
#pragma unroll
        for (int v = 0; v < 8; ++v)
          S[(rloc + v) * LDS_STRIDE + cloc] = c[v];
      }
    }
    __syncthreads();

    // ---- Stage B: scan own row, 8 candidates per guard ----
    // NOT unrolled: keeps the rare rescue/insert path in the code ONCE
    // (i-cache: unrolled x8 this loop body was ~20K instructions).
#pragma unroll 1
    for (int g = 0; g < CHUNK_COLS / 8; ++g) {
      const v4f q0 = Srow[2*g + 0];             // ds_load_b128
      const v4f q1 = Srow[2*g + 1];
      const float m0 = fminf(fminf(q0.x, q0.y), fminf(q0.z, q0.w));
      const float m1 = fminf(fminf(q1.x, q1.y), fminf(q1.z, q1.w));
      if (fminf(m0, m1) < best[KNN - 1]) {      // rare: some candidate qualifies
        const float vals[8] = {q0.x, q0.y, q0.z, q0.w, q1.x, q1.y, q1.z, q1.w};
#pragma unroll
        for (int u = 0; u < 8; ++u) {
          const float s = vals[u];
          if (s < best[KNN - 1]) {              // sorted bubble insert
            float v  = s;
            int   vi = j0 + g * 8 + u;
#pragma unroll
            for (int k = 0; k < KNN; ++k) {
              if (v < best[k]) {
                float tf = best[k]; best[k] = v;  v  = tf;
                int   ti = bidx[k]; bidx[k] = vi; vi = ti;
              }
            }
          }
        }
      }
    }
    __syncthreads();
  }

  // ---- Final Voronoi edge computation for this thread's point ----
  const int i = blockIdx.x * ROWS_PER_BLOCK + tid;
  const float px = points[3*i+0], py = points[3*i+1], pz = points[3*i+2];

  const int i0 = bidx[0];
  const float p0x = cells[3*i0+0], p0y = cells[3*i0+1], p0z = cells[3*i0+2];
  const float tcx = px - p0x, tcy = py - p0y, tcz = pz - p0z;

  float res = 3.4e38f;
#pragma unroll
  for (int e = 1; e < KNN; ++e) {
    const int ie = bidx[e];
    const float ex = cells[3*ie+0] - p0x;
    const float ey = cells[3*ie+1] - p0y;
    const float ez = cells[3*ie+2] - p0z;
    const float el2 = ex*ex + ey*ey + ez*ez;
    const float el  = sqrtf(el2);
    const float vl  = (tcx*ex + tcy*ey + tcz*ez) / el;
    const float d   = vl - 0.5f * el;
    res = fminf(res, d * d);
  }
  out[i] = res;
}

extern "C" void kernel_launch(void* const* d_in, const int* in_sizes, int n_in,
                              void* d_out, int out_size, void* d_ws, size_t ws_size,
                              hipStream_t stream) {
  (void)n_in; (void)d_ws; (void)ws_size; (void)out_size;
  const float* points = (const float*)d_in[0];   // (M,3) f32
  const float* cells  = (const float*)d_in[1];   // (N,3) f32
  float* out = (float*)d_out;                    // (M,) f32
  const int M = in_sizes[0] / 3;
  const int N = in_sizes[1] / 3;
  const int grid = M / ROWS_PER_BLOCK;           // 16384/128 = 128 blocks
  VoronoiValues_kernel<<<grid, THREADS, 0, stream>>>(points, cells, out, M, N);
}